// RXNEnc_781684048046
// MI455X (gfx1250) — compile-verified
//
#include <hip/hip_runtime.h>
#include <math.h>
#include <stdint.h>

typedef float v2f __attribute__((ext_vector_type(2)));
typedef float v8f __attribute__((ext_vector_type(8)));

#define B_GR   256
#define N_NODES 16384   // 256*64
#define N_EDGES 65536   // 256*256
#define N_SEQ   65536   // 256*256
#define OUTD    512
#define NH      4
#define ECHUNK  8192
#define BSTRIDE 72      // LDS row stride (floats); 2*72 % 64 == 16 -> half-waves hit disjoint banks

#if defined(__has_builtin)
#  if __has_builtin(__builtin_amdgcn_global_load_async_to_lds_b128) && \
      __has_builtin(__builtin_amdgcn_s_wait_asynccnt)
#    define HAVE_ASYNC_LDS 1
#  endif
#endif
#ifndef HAVE_ASYNC_LDS
#  define HAVE_ASYNC_LDS 0
#endif

#if HAVE_ASYNC_LDS
typedef int av4i __attribute__((vector_size(16)));                 // builtin param 0: av4i*
typedef __attribute__((address_space(3))) av4i av4i_lds;           // builtin param 1: av4i_lds*
#endif

__device__ inline v8f wmma4(v2f a, v2f b, v8f c)
{
  return __builtin_amdgcn_wmma_f32_16x16x4_f32(false, a, false, b, (short)0, c, false, false);
}

__device__ inline float silu_f(float v) { return v / (1.f + __expf(-v)); }

// ---------------------------------------------------------------------------
// fp32 GEMM on the WMMA pipe: C[M,N] = A[M,K] @ B[K,N] (+bias) (+silu)
// Block tile 128x64 (4 waves x 32 rows); B panel (16K x 64N) staged in LDS
// once per block (async global->LDS when available) and shared by all waves.
// LDS rows beyond K are zero-filled, so the compute loop needs no K masking:
// A addresses are merely clamped (real_value * 0 == 0). Only the N-edge path
// (last tile of N=852/1448) falls back to direct global loads with selects.
// Layouts (ISA 7.12.2, V_WMMA_F32_16X16X4_F32):
//   A: lane l, reg r -> A[l&15][k0 + 2*(l>>4) + r]
//   B: lane l, reg r -> B[k0 + 2*(l>>4) + r][n0 + (l&15)]
//   C: lane l, reg r -> C[m0 + r + 8*(l>>4)][n0 + (l&15)]
// ---------------------------------------------------------------------------
__global__ __launch_bounds__(128) void gemm_f32_wmma(
    const float* __restrict__ A, const float* __restrict__ Bm,
    const float* __restrict__ bias, float* __restrict__ C,
    int M, int N, int K, int act)
{
  __shared__ float Bt[16 * BSTRIDE];
  const int lane = threadIdx.x & 31;
  const int wv   = threadIdx.x >> 5;
  const int m0 = blockIdx.y * 128 + wv * 32;
  const int n0 = blockIdx.x * 64;
  const int lm = lane & 15;
  const int lk = (lane >> 4) << 1;     // 0 or 2
  const int rOff = (lane >> 4) * 8;
  v8f acc[8] = {};

  const bool fullN = (n0 + 64 <= N);   // uniform per block -> barriers stay uniform
  if (fullN) {
    // rows clamped; garbage rows only feed C rows that are never stored
    const int rA0 = min(m0 + lm, M - 1);
    const int rA1 = min(m0 + 16 + lm, M - 1);
    const float* Ar0 = A + (size_t)rA0 * K;
    const float* Ar1 = A + (size_t)rA1 * K;
    // cooperative B loader: 128 threads, 8 floats each (one 16x64 panel)
    const int lrow = threadIdx.x >> 3;         // 0..15
    const int lcol = (threadIdx.x & 7) * 8;    // 0,8,...,56
    float* Bdst = &Bt[lrow * BSTRIDE + lcol];

    for (int kc = 0; kc < K; kc += 16) {
      __syncthreads();                 // previous panel fully consumed
      const int kg = kc + lrow;
      if (kg < K) {
        const float* Bsrc = Bm + (size_t)kg * N + n0 + lcol;
#if HAVE_ASYNC_LDS
        __builtin_amdgcn_global_load_async_to_lds_b128(
            (av4i*)(uintptr_t)Bsrc,
            (av4i_lds*)(uintptr_t)(unsigned)(uintptr_t)Bdst, 0, 0);
        __builtin_amdgcn_global_load_async_to_lds_b128(
            (av4i*)(uintptr_t)(Bsrc + 4),
            (av4i_lds*)(uintptr_t)(unsigned)(uintptr_t)(Bdst + 4), 0, 0);
#else
#pragma unroll
        for (int j = 0; j < 8; ++j) Bdst[j] = Bsrc[j];
#endif
      } else {
#pragma unroll
        for (int j = 0; j < 8; ++j) Bdst[j] = 0.f;   // zero-pad K tail
      }
#if HAVE_ASYNC_LDS
      __builtin_amdgcn_s_wait_asynccnt(0);
#endif
      __syncthreads();                 // panel visible to all waves

      const int kend = min(16, K - kc);
      for (int k0 = 0; k0 < kend; k0 += 4) {
        const int kag = kc + k0 + lk;
        const int kg0 = min(kag, K - 1), kg1 = min(kag + 1, K - 1);
        v2f a0, a1;
        a0.x = Ar0[kg0]; a0.y = Ar0[kg1];
        a1.x = Ar1[kg0]; a1.y = Ar1[kg1];
        const float* Bl = &Bt[(k0 + lk) * BSTRIDE + lm];
#pragma unroll
        for (int t = 0; t < 4; ++t) {
          v2f b;
          b.x = Bl[t * 16];
          b.y = Bl[t * 16 + BSTRIDE];
          acc[t]     = wmma4(a0, b, acc[t]);
          acc[4 + t] = wmma4(a1, b, acc[4 + t]);
        }
      }
    }

    if (m0 + 32 <= M) {                // full-M epilogue: unconditional stores
#pragma unroll
      for (int t = 0; t < 4; ++t) {
        const int col = n0 + t * 16 + lm;
        const float bv = bias ? bias[col] : 0.f;
#pragma unroll
        for (int r = 0; r < 8; ++r) {
          float v0c = acc[t][r] + bv;
          float v1c = acc[4 + t][r] + bv;
          if (act == 1) { v0c = silu_f(v0c); v1c = silu_f(v1c); }
          C[(size_t)(m0 + rOff + r) * N + col]      = v0c;
          C[(size_t)(m0 + 16 + rOff + r) * N + col] = v1c;
        }
      }
    } else {                           // M edge (never hit for our sizes)
#pragma unroll
      for (int t = 0; t < 4; ++t) {
        const int col = n0 + t * 16 + lm;
        const float bv = bias ? bias[col] : 0.f;
#pragma unroll
        for (int r = 0; r < 8; ++r) {
          const int row0 = m0 + rOff + r, row1 = row0 + 16;
          if (row0 < M) {
            float v = acc[t][r] + bv;
            if (act == 1) v = silu_f(v);
            C[(size_t)row0 * N + col] = v;
          }
          if (row1 < M) {
            float v = acc[4 + t][r] + bv;
            if (act == 1) v = silu_f(v);
            C[(size_t)row1 * N + col] = v;
          }
        }
      }
    }
  } else {
    // ---- N-edge tile: direct global loads; A zeroed past K; B/rows clamped;
    // ---- garbage only reaches unstored C entries. No barriers here.
    if (m0 >= M) return;
    const int rA0 = min(m0 + lm, M - 1), rA1 = min(m0 + 16 + lm, M - 1);
    for (int k0 = 0; k0 < K; k0 += 4) {
      const int ka  = k0 + lk;
      const int kg0 = min(ka, K - 1), kg1 = min(ka + 1, K - 1);
      const bool v0 = ka < K, v1 = ka + 1 < K;
      float t00 = A[(size_t)rA0 * K + kg0], t01 = A[(size_t)rA0 * K + kg1];
      float t10 = A[(size_t)rA1 * K + kg0], t11 = A[(size_t)rA1 * K + kg1];
      v2f a0, a1;
      a0.x = v0 ? t00 : 0.f; a0.y = v1 ? t01 : 0.f;
      a1.x = v0 ? t10 : 0.f; a1.y = v1 ? t11 : 0.f;
#pragma unroll
      for (int t = 0; t < 4; ++t) {
        const int colC = min(n0 + t * 16 + lm, N - 1);
        v2f b;
        b.x = Bm[(size_t)kg0 * N + colC];
        b.y = Bm[(size_t)kg1 * N + colC];
        acc[t]     = wmma4(a0, b, acc[t]);
        acc[4 + t] = wmma4(a1, b, acc[4 + t]);
      }
    }
#pragma unroll
    for (int t = 0; t < 4; ++t) {
      const int col = n0 + t * 16 + lm;
      if (col >= N) continue;
      const float bv = bias ? bias[col] : 0.f;
#pragma unroll
      for (int r = 0; r < 8; ++r) {
        const int row0 = m0 + rOff + r, row1 = row0 + 16;
        if (row0 < M) {
          float v = acc[t][r] + bv;
          if (act == 1) v = silu_f(v);
          C[(size_t)row0 * N + col] = v;
        }
        if (row1 < M) {
          float v = acc[4 + t][r] + bv;
          if (act == 1) v = silu_f(v);
          C[(size_t)row1 * N + col] = v;
        }
      }
    }
  }
}

// ---------------------------------------------------------------------------
// Small helper kernels
// ---------------------------------------------------------------------------
__global__ void fill_kernel(float* p, float v, size_t n)
{
  size_t i = (size_t)blockIdx.x * blockDim.x + threadIdx.x;
  size_t stride = (size_t)gridDim.x * blockDim.x;
  for (; i < n; i += stride) p[i] = v;
}

__global__ void embed_kernel(const int* __restrict__ x, const float* __restrict__ emb,
                             float* __restrict__ h)
{
  int i = blockIdx.x * blockDim.x + threadIdx.x;   // over N_NODES*64
  if (i >= N_NODES * 64) return;
  int n = i >> 6, d = i & 63;
  h[i] = emb[x[n] * 64 + d];
}

__device__ inline void atomicMaxF(float* addr, float v)
{
  if (v >= 0.f) atomicMax((int*)addr, __float_as_int(v));
  else          atomicMin((unsigned int*)addr, __float_as_uint(v));
}

// wave per (edge, head): alpha = att_h . leaky_relu(xl[src]+xr[dst]+em, 0.2)
__global__ void gat_alpha_kernel(const float* __restrict__ xl, const float* __restrict__ xr,
                                 const float* __restrict__ em, const int* __restrict__ src,
                                 const int* __restrict__ dst, const float* __restrict__ att,
                                 const float* __restrict__ evalid,
                                 float* __restrict__ alpha, float* __restrict__ amax,
                                 int e_begin, int e_count, int do_, int HD)
{
  int gw = (blockIdx.x * blockDim.x + threadIdx.x) >> 5;
  int lane = threadIdx.x & 31;
  if (gw >= e_count * NH) return;
  int le = gw >> 2, hh = gw & 3;
  int e = e_begin + le;
  int s = src[e], d = dst[e];
  const float* xls = xl + (size_t)s * HD + hh * do_;
  const float* xrd = xr + (size_t)d * HD + hh * do_;
  const float* emr = em + (size_t)le * HD + hh * do_;
  const float* at  = att + hh * do_;
  float sum = 0.f;
  for (int j = lane; j < do_; j += 32) {
    float m = xls[j] + xrd[j] + emr[j];
    m = (m > 0.f) ? m : 0.2f * m;
    sum += at[j] * m;
  }
  for (int off = 16; off > 0; off >>= 1) sum += __shfl_down(sum, off, 32);
  if (lane == 0) {
    float ev = evalid[e];
    alpha[e * NH + hh] = (ev > 0.f) ? sum : __int_as_float(0xff800000u);
    if (ev > 0.f) atomicMaxF(&amax[d * NH + hh], sum);
  }
}

__global__ void gat_softmax_kernel(const float* __restrict__ alpha, const int* __restrict__ dst,
                                   const float* __restrict__ amax, const float* __restrict__ evalid,
                                   float* __restrict__ exw, float* __restrict__ den)
{
  int i = blockIdx.x * blockDim.x + threadIdx.x;   // over E*NH
  if (i >= N_EDGES * NH) return;
  int e = i >> 2, hh = i & 3;
  int d = dst[e];
  float am = amax[d * NH + hh];
  if (!(am > -1e37f)) am = 0.f;                    // isfinite(amax) ? amax : 0
  float a = alpha[i];
  float ex = (evalid[e] > 0.f && a > -1e37f) ? __expf(a - am) : 0.f;
  exw[i] = ex;
  if (ex != 0.f) atomicAdd(&den[d * NH + hh], ex);
}

// grid(E, ceil(do/256)); accumulate head-mean message into hn[dst]
__global__ void gat_aggregate_kernel(const float* __restrict__ xl, const int* __restrict__ src,
                                     const int* __restrict__ dst, const float* __restrict__ exw,
                                     const float* __restrict__ den, float* __restrict__ hn,
                                     int do_, int HD)
{
  int e = blockIdx.x;
  int j = blockIdx.y * blockDim.x + threadIdx.x;
  if (j >= do_) return;
  int s = src[e], d = dst[e];
  float acc = 0.f;
#pragma unroll
  for (int hh = 0; hh < NH; ++hh) {
    float w = exw[e * NH + hh];
    if (w != 0.f) {
      float dn = den[d * NH + hh];
      acc += (w / fmaxf(dn, 1e-16f)) * xl[(size_t)s * HD + hh * do_ + j];
    }
  }
  if (acc != 0.f) atomicAdd(&hn[(size_t)d * do_ + j], 0.25f * acc);
}

__global__ void gat_finish_kernel(const float* __restrict__ hn, const float* __restrict__ bias,
                                  const float* __restrict__ nvalid, float* __restrict__ h, int do_)
{
  int n = blockIdx.x;
  int j = blockIdx.y * blockDim.x + threadIdx.x;
  if (j >= do_) return;
  h[(size_t)n * do_ + j] = (hn[(size_t)n * do_ + j] + bias[j]) * nvalid[n];
}

// per-graph mean pool over 64 contiguous nodes (h already masked)
__global__ void pool_mean_kernel(const float* __restrict__ h, const float* __restrict__ nvalid,
                                 float* __restrict__ pooled, int do_)
{
  int b = blockIdx.x, t = threadIdx.x;
  __shared__ float red[256];
  red[t] = (t < 64) ? nvalid[b * 64 + t] : 0.f;
  __syncthreads();
  for (int off = 128; off > 0; off >>= 1) {
    if (t < off) red[t] += red[t + off];
    __syncthreads();
  }
  float cnt = fmaxf(red[0], 1.f);
  for (int ch = t; ch < do_; ch += 256) {
    float s = 0.f;
    for (int j = 0; j < 64; ++j) s += h[(size_t)(b * 64 + j) * do_ + ch];
    pooled[b * do_ + ch] = s / cnt;
  }
}

// row LayerNorm + SiLU; addTo: Y += result else Y = result
__global__ void ln_silu_kernel(const float* __restrict__ X, const float* __restrict__ g,
                               const float* __restrict__ be, float* __restrict__ Y,
                               int D, int addTo)
{
  int row = blockIdx.x, t = threadIdx.x;
  const float* x = X + (size_t)row * D;
  float s = 0.f, s2 = 0.f;
  for (int j = t; j < D; j += 256) { float v = x[j]; s += v; s2 += v * v; }
  __shared__ float rs[256], rs2[256];
  rs[t] = s; rs2[t] = s2; __syncthreads();
  for (int off = 128; off > 0; off >>= 1) {
    if (t < off) { rs[t] += rs[t + off]; rs2[t] += rs2[t + off]; }
    __syncthreads();
  }
  float mu = rs[0] / D;
  float inv = rsqrtf(fmaxf(rs2[0] / D - mu * mu, 0.f) + 1e-5f);
  for (int j = t; j < D; j += 256) {
    float v = (x[j] - mu) * inv * g[j] + be[j];
    v = silu_f(v);
    if (addTo) Y[(size_t)row * D + j] += v; else Y[(size_t)row * D + j] = v;
  }
}

__global__ void lrelu_kernel(float* __restrict__ h, float slope, size_t n)
{
  size_t i = (size_t)blockIdx.x * blockDim.x + threadIdx.x;
  if (i >= n) return;
  float v = h[i];
  h[i] = (v > 0.f) ? v : slope * v;
}

__global__ void sag_agg_kernel(const float* __restrict__ h, const int* __restrict__ src,
                               const int* __restrict__ dst, const float* __restrict__ evalid,
                               float* __restrict__ agg, int do_)
{
  int e = blockIdx.x;
  if (evalid[e] <= 0.f) return;
  int j = blockIdx.y * blockDim.x + threadIdx.x;
  if (j >= do_) return;
  float v = h[(size_t)src[e] * do_ + j];
  if (v != 0.f) atomicAdd(&agg[(size_t)dst[e] * do_ + j], v);
}

__global__ void sag_score_kernel(const float* __restrict__ agg, const float* __restrict__ h,
                                 const float* __restrict__ Wrel, const float* __restrict__ brel,
                                 const float* __restrict__ Wroot, const float* __restrict__ nvalid,
                                 float* __restrict__ score, int do_)
{
  int gw = (blockIdx.x * blockDim.x + threadIdx.x) >> 5;
  int lane = threadIdx.x & 31;
  if (gw >= N_NODES) return;
  const float* ag = agg + (size_t)gw * do_;
  const float* hr = h + (size_t)gw * do_;
  float s = 0.f;
  for (int j = lane; j < do_; j += 32) s += ag[j] * Wrel[j] + hr[j] * Wroot[j];
  for (int off = 16; off > 0; off >>= 1) s += __shfl_down(s, off, 32);
  if (lane == 0) {
    s += brel[0];
    score[gw] = (nvalid[gw] > 0.f) ? s : __int_as_float(0xff800000u);
  }
}

// 64 threads per graph: stable top-k by (score desc, index asc)
__global__ void topk_gate_kernel(const float* __restrict__ score, float* __restrict__ nvalid,
                                 float* __restrict__ gatev, int k)
{
  int b = blockIdx.x, j = threadIdx.x;
  __shared__ float sc[64];
  float s = score[b * 64 + j];
  sc[j] = s;
  __syncthreads();
  int rank = 0;
  for (int t = 0; t < 64; ++t) {
    float o = sc[t];
    if (o > s || (o == s && t < j)) rank++;
  }
  bool keep = rank < k;
  nvalid[b * 64 + j] = keep ? 1.f : 0.f;
  gatev[b * 64 + j] = keep ? tanhf(s) : 0.f;
}

__global__ void scale_h_kernel(float* __restrict__ h, const float* __restrict__ gatev, int do_)
{
  int n = blockIdx.x;
  int j = blockIdx.y * blockDim.x + threadIdx.x;
  if (j >= do_) return;
  h[(size_t)n * do_ + j] *= gatev[n];
}

__global__ void edge_update_kernel(const int* __restrict__ src, const int* __restrict__ dst,
                                   const float* __restrict__ nvalid, float* __restrict__ evalid)
{
  int e = blockIdx.x * blockDim.x + threadIdx.x;
  if (e >= N_EDGES) return;
  evalid[e] *= nvalid[src[e]] * nvalid[dst[e]];
}

// per-graph softmax-gated pooling over 256 contiguous tokens
__global__ void seq_attn_pool_kernel(const float* __restrict__ seqh, const float* __restrict__ gate,
                                     float* __restrict__ out)
{
  int b = blockIdx.x, t = threadIdx.x;
  for (int c = t; c < 512; c += 256) {
    const float* gb = gate + (size_t)b * 256 * 512 + c;
    const float* hb = seqh + (size_t)b * 256 * 512 + c;
    float mx = -3.0e38f;
    for (int s2 = 0; s2 < 256; ++s2) mx = fmaxf(mx, gb[(size_t)s2 * 512]);
    float dn = 0.f, num = 0.f;
    for (int s2 = 0; s2 < 256; ++s2) {
      float ex = __expf(gb[(size_t)s2 * 512] - mx);
      dn += ex;
      num += ex * hb[(size_t)s2 * 512];
    }
    out[b * 512 + c] = num / dn;
  }
}

__global__ void add3_kernel(const float* __restrict__ a, const float* __restrict__ b,
                            const float* __restrict__ c, float* __restrict__ o, int n)
{
  int i = blockIdx.x * blockDim.x + threadIdx.x;
  if (i >= n) return;
  o[i] = a[i] + b[i] + c[i];
}

// ---------------------------------------------------------------------------
extern "C" void kernel_launch(void* const* d_in, const int* in_sizes, int n_in,
                              void* d_out, int out_size, void* d_ws, size_t ws_size,
                              hipStream_t stream)
{
  (void)in_sizes; (void)n_in; (void)out_size; (void)ws_size;
  const int*   x       = (const int*)  d_in[0];
  const int*   srcI    = (const int*)  d_in[1];
  const int*   dstI    = srcI + N_EDGES;
  const float* edgeAt  = (const float*)d_in[2];
  const float* drfp    = (const float*)d_in[3];
  const float* seq     = (const float*)d_in[4];
  const float* atomEmb = (const float*)d_in[7];
  const float* edgeW   = (const float*)d_in[8];
  const float* edgeB   = (const float*)d_in[9];
  const float* drfpW   = (const float*)d_in[10];
  const float* drfpB   = (const float*)d_in[11];
  const float* seqW    = (const float*)d_in[12];
  const float* seqB    = (const float*)d_in[13];
  const float* seqG    = (const float*)d_in[14];
  const float* seqBe   = (const float*)d_in[15];
  const float* gateW   = (const float*)d_in[16];
  const float* gateB   = (const float*)d_in[17];
  const float* gateG   = (const float*)d_in[18];
  const float* gateBe  = (const float*)d_in[19];
  const float* fcW     = (const float*)d_in[20];
  const float* fcB     = (const float*)d_in[21];
  const float *Wl[3], *Wr[3], *We[3], *att[3], *gbias[3];
  for (int i = 0; i < 3; ++i) {
    Wl[i]    = (const float*)d_in[22 + 5 * i];
    Wr[i]    = (const float*)d_in[23 + 5 * i];
    We[i]    = (const float*)d_in[24 + 5 * i];
    att[i]   = (const float*)d_in[25 + 5 * i];
    gbias[i] = (const float*)d_in[26 + 5 * i];
  }
  const float *mW[3], *mB[3], *mG[3], *mBe[3];
  for (int i = 0; i < 3; ++i) {
    mW[i]  = (const float*)d_in[37 + 4 * i];
    mB[i]  = (const float*)d_in[38 + 4 * i];
    mG[i]  = (const float*)d_in[39 + 4 * i];
    mBe[i] = (const float*)d_in[40 + 4 * i];
  }
  const float *pWrel[2], *pBrel[2], *pWroot[2];
  for (int i = 0; i < 2; ++i) {
    pWrel[i]  = (const float*)d_in[49 + 3 * i];
    pBrel[i]  = (const float*)d_in[50 + 3 * i];
    pWroot[i] = (const float*)d_in[51 + 3 * i];
  }

  // ---- workspace bump allocation (floats) ----
  float* wsf = (float*)d_ws;
  size_t off = 0;
  auto A = [&](size_t n) { float* p = wsf + off; off += n; return p; };
  float* e_buf   = A((size_t)N_EDGES * 64);
  float* h_buf   = A((size_t)N_NODES * 512);
  float* hn_buf  = A((size_t)N_NODES * 512);
  float* xl_buf  = A((size_t)N_NODES * 2048);
  float* xr_buf  = A((size_t)N_NODES * 2048);
  float* em_buf  = A((size_t)ECHUNK * 2048);
  float* alpha_b = A((size_t)N_EDGES * NH);
  float* exw_b   = A((size_t)N_EDGES * NH);
  float* amax_b  = A((size_t)N_NODES * NH);
  float* den_b   = A((size_t)N_NODES * NH);
  float* score_b = A(N_NODES);
  float* gate_v  = A(N_NODES);
  float* nvalid  = A(N_NODES);
  float* evalid  = A(N_EDGES);
  float* pooled  = A(256 * 512);
  float* tmp_b   = A(256 * 512);
  float* gout    = A(256 * 512);
  float* sout    = A(256 * 512);
  float* sum_b   = A(256 * 512);
  float* agg_buf = hn_buf;   // reuse
  float* seqh    = xl_buf;   // reuse after graph loop
  float* gate_b2 = xr_buf;   // reuse after graph loop

  auto gemm = [&](const float* Am, const float* Bm, const float* bias, float* Cm,
                  int M, int N, int K, int act) {
    dim3 g((N + 63) / 64, (M + 127) / 128);
    gemm_f32_wmma<<<g, 128, 0, stream>>>(Am, Bm, bias, Cm, M, N, K, act);
  };
  auto fill = [&](float* p, float v, size_t n) {
    int blocks = (int)((n + 255) / 256);
    fill_kernel<<<blocks, 256, 0, stream>>>(p, v, n);
  };

  const int DI[3] = {64, 213, 362};
  const int DO[3] = {213, 362, 512};

  fill(gout, 0.f, 256 * 512);
  fill(nvalid, 1.f, N_NODES);
  fill(evalid, 1.f, N_EDGES);

  // node embeddings + edge feature linear (K=18)
  embed_kernel<<<(N_NODES * 64 + 255) / 256, 256, 0, stream>>>(x, atomEmb, h_buf);
  gemm(edgeAt, edgeW, edgeB, e_buf, N_EDGES, 64, 18, 0);

  int kk = 64;
  for (int i = 0; i < 3; ++i) {
    const int di = DI[i], do_ = DO[i], HD = NH * do_;
    // source/target transforms
    gemm(h_buf, Wl[i], nullptr, xl_buf, N_NODES, HD, di, 0);
    gemm(h_buf, Wr[i], nullptr, xr_buf, N_NODES, HD, di, 0);
    fill(amax_b, -INFINITY, (size_t)N_NODES * NH);
    // em = e @ We in chunks, fused immediately into attention logits
    for (int c = 0; c < N_EDGES / ECHUNK; ++c) {
      gemm(e_buf + (size_t)c * ECHUNK * 64, We[i], nullptr, em_buf, ECHUNK, HD, 64, 0);
      int waves = ECHUNK * NH;
      gat_alpha_kernel<<<(waves * 32 + 255) / 256, 256, 0, stream>>>(
          xl_buf, xr_buf, em_buf, srcI, dstI, att[i], evalid, alpha_b, amax_b,
          c * ECHUNK, ECHUNK, do_, HD);
    }
    fill(den_b, 0.f, (size_t)N_NODES * NH);
    gat_softmax_kernel<<<(N_EDGES * NH + 255) / 256, 256, 0, stream>>>(
        alpha_b, dstI, amax_b, evalid, exw_b, den_b);
    fill(hn_buf, 0.f, (size_t)N_NODES * do_);
    gat_aggregate_kernel<<<dim3(N_EDGES, (do_ + 255) / 256), 256, 0, stream>>>(
        xl_buf, srcI, dstI, exw_b, den_b, hn_buf, do_, HD);
    gat_finish_kernel<<<dim3(N_NODES, (do_ + 255) / 256), 256, 0, stream>>>(
        hn_buf, gbias[i], nvalid, h_buf, do_);
    // graph readout + MLP (Linear -> LN -> SiLU), accumulated into gout
    pool_mean_kernel<<<B_GR, 256, 0, stream>>>(h_buf, nvalid, pooled, do_);
    gemm(pooled, mW[i], mB[i], tmp_b, B_GR, OUTD, do_, 0);
    ln_silu_kernel<<<B_GR, 256, 0, stream>>>(tmp_b, mG[i], mBe[i], gout, OUTD, 1);
    if (i < 2) {
      lrelu_kernel<<<(int)(((size_t)N_NODES * do_ + 255) / 256), 256, 0, stream>>>(
          h_buf, 0.01f, (size_t)N_NODES * do_);
      fill(agg_buf, 0.f, (size_t)N_NODES * do_);
      sag_agg_kernel<<<dim3(N_EDGES, (do_ + 255) / 256), 256, 0, stream>>>(
          h_buf, srcI, dstI, evalid, agg_buf, do_);
      sag_score_kernel<<<(N_NODES * 32 + 255) / 256, 256, 0, stream>>>(
          agg_buf, h_buf, pWrel[i], pBrel[i], pWroot[i], nvalid, score_b, do_);
      kk = (kk + 1) / 2;
      topk_gate_kernel<<<B_GR, 64, 0, stream>>>(score_b, nvalid, gate_v, kk);
      scale_h_kernel<<<dim3(N_NODES, (do_ + 255) / 256), 256, 0, stream>>>(h_buf, gate_v, do_);
      edge_update_kernel<<<(N_EDGES + 255) / 256, 256, 0, stream>>>(srcI, dstI, nvalid, evalid);
    }
  }

  // sequence branch: Dense(768->512)+LN+SiLU, gate Dense(512->512)+LN+SiLU, attn pool
  gemm(seq, seqW, seqB, seqh, N_SEQ, OUTD, 768, 0);
  ln_silu_kernel<<<N_SEQ, 256, 0, stream>>>(seqh, seqG, seqBe, seqh, OUTD, 0);
  gemm(seqh, gateW, gateB, gate_b2, N_SEQ, OUTD, OUTD, 0);
  ln_silu_kernel<<<N_SEQ, 256, 0, stream>>>(gate_b2, gateG, gateBe, gate_b2, OUTD, 0);
  seq_attn_pool_kernel<<<B_GR, 256, 0, stream>>>(seqh, gate_b2, sout);

  // drfp branch + combine + final FC (Linear -> SiLU) straight into d_out
  gemm(drfp, drfpW, drfpB, tmp_b, B_GR, OUTD, 2048, 0);
  add3_kernel<<<(B_GR * OUTD + 255) / 256, 256, 0, stream>>>(gout, sout, tmp_b, sum_b, B_GR * OUTD);
  gemm(sum_b, fcW, fcB, (float*)d_out, B_GR, OUTD, OUTD, 1);
}